// TwoLayerGCN_5342939316957
// MI455X (gfx1250) — compile-verified
//
#include <hip/hip_runtime.h>
#include <hip/hip_bf16.h>
#include <math.h>

typedef __attribute__((ext_vector_type(2))) float v2f;
typedef __attribute__((ext_vector_type(8))) float v8f;

__device__ __forceinline__ void atomAddF32(float* p, float v) {
  unsafeAtomicAdd(p, v);   // hardware global_atomic_add_f32
}

// ---------------- zero fill ----------------
__global__ void zero_f32(float* __restrict__ p, size_t n) {
  size_t i = (size_t)blockIdx.x * blockDim.x + threadIdx.x;
  size_t stride = (size_t)gridDim.x * blockDim.x;
  for (; i < n; i += stride) p[i] = 0.0f;
}

// ---------------- GEMM: C[Mx256] = act(A[MxK]) @ B[Kx256] ----------------
// fp32 WMMA 16x16x4. Block = 32x8 (8 waves).
// waves 0-3 -> rows [blk*32, +16), n-groups 0..3 (64 cols each)
// waves 4-7 -> rows [blk*32+16, +16)
//
// B staged in K-pair-interleaved LDS layout: Bs[p*544 + 2n + q] = B[kb+2p+q][n]
//   -> each B fragment is ONE contiguous ds_load_b64 into an even VGPR pair
//   -> pair stride 544 (mod 64 = 32): lanes 0-15 and 16-31 hit disjoint bank halves
// A: GEMM1 (no ReLU) staged via async global->LDS DMA (ASYNCcnt path);
//    GEMM2 staged via VGPRs with fused ReLU.
template<int K, bool RELU_A>
__global__ __launch_bounds__(256) void gemm_wmma(const float* __restrict__ A,
                                                 const float* __restrict__ B,
                                                 float* __restrict__ C) {
  constexpr int KC  = 32;          // K chunk (divides 768 and 256)
  constexpr int LDA = KC + 4;      // 36: disjoint bank pairs for A b64 frag loads
  constexpr int PST = 512 + 32;    // 544: pair stride (floats)
  __shared__ float As[32 * LDA];        // 4608 B
  __shared__ float Bs[(KC / 2) * PST];  // 16*544*4 = 34816 B

  const int lane  = threadIdx.x;            // 0..31
  const int wave  = threadIdx.y;            // 0..7
  const int tid   = wave * 32 + lane;       // 0..255
  const int mt    = wave >> 2;              // mtile within block (0/1)
  const int mbase = blockIdx.x * 32;
  const int m0    = mbase + mt * 16;
  const int n0    = (wave & 3) * 64;
  const int r16   = lane & 15;
  const int koff  = (lane < 16) ? 0 : 2;

  // A staging: exactly one float4 per thread: row = tid>>3, kq = (tid&7)*4
  const int arow = tid >> 3;
  const int akq  = (tid & 7) << 2;
  const float* Aptr  = A + (size_t)(mbase + arow) * K + akq;
  float*       AsDst = &As[arow * LDA + akq];
  const uint32_t AsOff = (uint32_t)(uintptr_t)AsDst;

  v8f acc0 = {}, acc1 = {}, acc2 = {}, acc3 = {};

  for (int kb = 0; kb < K; kb += KC) {
    __syncthreads();   // previous chunk's compute done before LDS overwrite

    // ---- stage A chunk ----
    if constexpr (RELU_A) {
      float4 s = *(const float4*)(Aptr + kb);
      s.x = fmaxf(s.x, 0.0f); s.y = fmaxf(s.y, 0.0f);
      s.z = fmaxf(s.z, 0.0f); s.w = fmaxf(s.w, 0.0f);
      *(float4*)AsDst = s;
    } else {
      // async DMA: 16 B per thread, bypasses VGPRs, tracked by ASYNCcnt
      asm volatile("global_load_async_to_lds_b128 %0, %1, off"
                   :: "v"(AsOff), "v"(Aptr + kb) : "memory");
    }

    // ---- stage B chunk (interleaved pairs): 2048 tasks, 8 per thread ----
    #pragma unroll
    for (int i = 0; i < 8; ++i) {
      const int task = tid + (i << 8);     // 0..2047
      const int p    = task >> 7;          // K pair 0..15
      const int n    = (task & 127) << 1;  // column 0,2,...,254
      const float* g0 = B + (size_t)(kb + 2 * p) * 256 + n;
      const v2f r0 = *(const v2f*)g0;          // B[2p  ][n..n+1]
      const v2f r1 = *(const v2f*)(g0 + 256);  // B[2p+1][n..n+1]
      *(float4*)(&Bs[p * PST + 2 * n]) = make_float4(r0.x, r1.x, r0.y, r1.y);
    }

    if constexpr (!RELU_A)
      asm volatile("s_wait_asynccnt 0x0" ::: "memory");
    __syncthreads();   // all staging visible to all 8 waves

    const float* Asw = &As[(mt * 16 + r16) * LDA + koff];
    // lane's B base: pair (koff/2), interleaved column slot 2*(n0+r16)
    const float* Bsw = &Bs[(koff >> 1) * PST + 2 * (n0 + r16)];
    #pragma unroll
    for (int k = 0; k < KC; k += 4) {
      v2f a = *(const v2f*)(Asw + k);           // ds_load_b64
      const float* Bk = Bsw + (k >> 1) * PST;   // pair k/2 (+1 for upper lanes)
      v2f b0 = *(const v2f*)(Bk +  0);          // ds_load_b64, contiguous pair
      v2f b1 = *(const v2f*)(Bk + 32);
      v2f b2 = *(const v2f*)(Bk + 64);
      v2f b3 = *(const v2f*)(Bk + 96);
      acc0 = __builtin_amdgcn_wmma_f32_16x16x4_f32(false, a, false, b0, (short)0, acc0, false, false);
      acc1 = __builtin_amdgcn_wmma_f32_16x16x4_f32(false, a, false, b1, (short)0, acc1, false, false);
      acc2 = __builtin_amdgcn_wmma_f32_16x16x4_f32(false, a, false, b2, (short)0, acc2, false, false);
      acc3 = __builtin_amdgcn_wmma_f32_16x16x4_f32(false, a, false, b3, (short)0, acc3, false, false);
    }
  }

  // C/D layout: VGPR r -> lanes 0-15: row m0+r ; lanes 16-31: row m0+8+r
  const int rowbase = m0 + ((lane < 16) ? 0 : 8);
  #pragma unroll
  for (int r = 0; r < 8; ++r) {
    size_t off = (size_t)(rowbase + r) * 256 + n0 + r16;
    C[off +  0] = acc0[r];
    C[off + 16] = acc1[r];
    C[off + 32] = acc2[r];
    C[off + 48] = acc3[r];
  }
}

// ---------------- COO SpMM scatter: Hout[rows[e]] += vals[e]*Hin[cols[e]] ----------------
__global__ __launch_bounds__(256) void spmm_scatter(const int* __restrict__ rows,
                                                    const int* __restrict__ cols,
                                                    const float* __restrict__ vals,
                                                    const float* __restrict__ Hin,
                                                    float* __restrict__ Hout, int E) {
  const int t   = threadIdx.x;
  const int c4  = (t & 63) * 4;   // 64 threads cover 256 cols with float4
  const int sub = t >> 6;         // 4 edge streams per block
  for (int e = blockIdx.x * 4 + sub; e < E; e += gridDim.x * 4) {
    const int r = rows[e], c = cols[e];
    const float v = vals[e];
    const float4 h = *(const float4*)(Hin + (size_t)c * 256 + c4);
    float* o = Hout + (size_t)r * 256 + c4;
    atomAddF32(o + 0, v * h.x);
    atomAddF32(o + 1, v * h.y);
    atomAddF32(o + 2, v * h.z);
    atomAddF32(o + 3, v * h.w);
  }
}

// ---------------- fused spmm2 + column pooling: pooled[j] = sum_e vals[e]*H[cols[e]][j] ----------------
__global__ __launch_bounds__(256) void edge_reduce(const int* __restrict__ cols,
                                                   const float* __restrict__ vals,
                                                   const float* __restrict__ H,
                                                   float* __restrict__ pooled, int E) {
  const int t   = threadIdx.x;
  const int c4  = (t & 63) * 4;
  const int sub = t >> 6;
  float a0 = 0.f, a1 = 0.f, a2 = 0.f, a3 = 0.f;
  for (int e = blockIdx.x * 4 + sub; e < E; e += gridDim.x * 4) {
    const int c = cols[e];
    const float v = vals[e];
    const float4 h = *(const float4*)(H + (size_t)c * 256 + c4);
    a0 += v * h.x; a1 += v * h.y; a2 += v * h.z; a3 += v * h.w;
  }
  atomAddF32(&pooled[c4 + 0], a0);
  atomAddF32(&pooled[c4 + 1], a1);
  atomAddF32(&pooled[c4 + 2], a2);
  atomAddF32(&pooled[c4 + 3], a3);
}

// ---------------- finalize: sigmoid -> concat cls -> [1024]x[1024,4] matvec + bias ----------------
__global__ __launch_bounds__(256) void finalize(const float* __restrict__ pooled,
                                                const float* __restrict__ cls,
                                                const float* __restrict__ W3,
                                                const float* __restrict__ b3,
                                                float* __restrict__ out) {
  __shared__ float red[256][4];
  const int t = threadIdx.x;
  float a0 = 0.f, a1 = 0.f, a2 = 0.f, a3 = 0.f;
  for (int i = t; i < 1024; i += 256) {
    float xv = (i < 256) ? (1.0f / (1.0f + expf(-pooled[i]))) : cls[i - 256];
    const float* w = W3 + (size_t)i * 4;
    a0 += xv * w[0]; a1 += xv * w[1]; a2 += xv * w[2]; a3 += xv * w[3];
  }
  red[t][0] = a0; red[t][1] = a1; red[t][2] = a2; red[t][3] = a3;
  __syncthreads();
  for (int s = 128; s > 0; s >>= 1) {
    if (t < s) {
      red[t][0] += red[t + s][0]; red[t][1] += red[t + s][1];
      red[t][2] += red[t + s][2]; red[t][3] += red[t + s][3];
    }
    __syncthreads();
  }
  if (t < 4) out[t] = red[0][t] + b3[t];
}

extern "C" void kernel_launch(void* const* d_in, const int* in_sizes, int n_in,
                              void* d_out, int out_size, void* d_ws, size_t ws_size,
                              hipStream_t stream) {
  const float* x    = (const float*)d_in[0];
  const int*   rows = (const int*)d_in[1];
  const int*   cols = (const int*)d_in[2];
  const float* vals = (const float*)d_in[3];
  const float* cls  = (const float*)d_in[4];
  const float* W1   = (const float*)d_in[5];
  const float* W2   = (const float*)d_in[6];
  const float* W3   = (const float*)d_in[7];
  const float* b3   = (const float*)d_in[8];
  float* out = (float*)d_out;

  const int N = in_sizes[0] / 768;   // 100000 (divisible by 32)
  const int E = in_sizes[1];         // 3200000

  float* bufA   = (float*)d_ws;                  // N x 256
  float* bufB   = bufA + (size_t)N * 256;        // N x 256
  float* pooled = bufB + (size_t)N * 256;        // 256

  // bufB + pooled are contiguous -> single zero pass
  zero_f32<<<2048, 256, 0, stream>>>(bufB, (size_t)N * 256 + 256);

  // t1 = x @ W1
  gemm_wmma<768, false><<<N / 32, dim3(32, 8), 0, stream>>>(x, W1, bufA);
  // h1 = spmm(A, t1)  (relu deferred into next GEMM's A-load)
  spmm_scatter<<<8192, 256, 0, stream>>>(rows, cols, vals, bufA, bufB, E);
  // t2 = relu(h1) @ W2
  gemm_wmma<256, true><<<N / 32, dim3(32, 8), 0, stream>>>(bufB, W2, bufA);
  // pooled = colsum(spmm(A, t2)) collapses to a single edge reduction
  edge_reduce<<<4096, 256, 0, stream>>>(cols, vals, bufA, pooled, E);
  // out = [sigmoid(pooled), cls] @ W3 + b3
  finalize<<<1, 256, 0, stream>>>(pooled, cls, W3, b3, out);
}